// BatchedExpertRouter_57226144252167
// MI455X (gfx1250) — compile-verified
//
#include <hip/hip_runtime.h>
#include <hip/hip_bf16.h>

// ---------------------------------------------------------------------------
// MoE token router (BatchedExpertRouter) for MI455X / gfx1250.
//
// Bandwidth-bound problem: dominant cost is the permuted copy of the
// [16384 x 4096] f32 token matrix (256 MB read + 256 MB write ~ 23 us at
// 23.3 TB/s). No matrix math -> no WMMA. CDNA5-specific path used: async
// global<->LDS DMA (ASYNCcnt) for the bulk row gather.
// ---------------------------------------------------------------------------

#define NTOK  16384
#define DIM   4096
#define NEXP  64
#define NPAT  4096      // NEXP^TOP_K = 64^2
#define TPB   256
#define NBLK  64        // NTOK / TPB

// d_out layout (floats), reference tuple return order
#define O_KEYS  ((size_t)0)
#define O_ORDER ((size_t)NTOK)
#define O_TOK   ((size_t)(2 * NTOK))
#define O_W     (O_TOK + (size_t)NTOK * DIM)
#define O_CNT   (O_W + (size_t)NTOK * 2)

// ---------------------------------------------------------------------------
__global__ __launch_bounds__(TPB) void zero_i32(int* __restrict__ p, int n) {
    int i = blockIdx.x * TPB + threadIdx.x;
    if (i < n) p[i] = 0;
}

// ---------------------------------------------------------------------------
// K1: per-token pattern key + per-block histogram (bin-major: hist[key*NBLK+b])
__global__ __launch_bounds__(TPB) void keys_hist(const int* __restrict__ eidx,
                                                 int* __restrict__ keys,
                                                 int* __restrict__ hist) {
    const int b = blockIdx.x, t = threadIdx.x;
    const int i = b * TPB + t;
    __shared__ int h[NPAT];
#pragma unroll
    for (int j = 0; j < NPAT / TPB; ++j) h[t + j * TPB] = 0;
    __syncthreads();

    int e0 = eidx[i * 2 + 0];
    int e1 = eidx[i * 2 + 1];
    int lo = e0 < e1 ? e0 : e1;
    int hi = e0 < e1 ? e1 : e0;
    int key = lo * NEXP + hi;
    keys[i] = key;
    atomicAdd(&h[key], 1);
    __syncthreads();

#pragma unroll
    for (int j = 0; j < NPAT / TPB; ++j) {
        int k = t + j * TPB;
        hist[k * NBLK + b] = h[k];
    }
}

// ---------------------------------------------------------------------------
// K2: single-workgroup scan. Per bin: exclusive scan of its 64 block counts
// (in place) + bin total. Exclusive scan over 4096 bin totals -> binBase.
// Also emits counts output (as float).
__global__ __launch_bounds__(TPB) void scan_bins(int* __restrict__ hist,
                                                 int* __restrict__ binBase,
                                                 float* __restrict__ countsOut) {
    const int t = threadIdx.x;           // 256 threads, 16 bins each
    __shared__ int partial[TPB];
    int tot[NPAT / TPB];
    int tsum = 0;
#pragma unroll
    for (int j = 0; j < NPAT / TPB; ++j) {
        const int k = t * (NPAT / TPB) + j;
        int run = 0;
        for (int b = 0; b < NBLK; ++b) {
            int c = hist[k * NBLK + b];
            hist[k * NBLK + b] = run;    // exclusive prefix over blocks
            run += c;
        }
        tot[j] = run;
        countsOut[k] = (float)run;
        tsum += run;
    }
    partial[t] = tsum;
    __syncthreads();
    if (t == 0) {
        int run = 0;
        for (int i = 0; i < TPB; ++i) { int c = partial[i]; partial[i] = run; run += c; }
    }
    __syncthreads();
    int base = partial[t];
#pragma unroll
    for (int j = 0; j < NPAT / TPB; ++j) {
        const int k = t * (NPAT / TPB) + j;
        binBase[k] = base;
        base += tot[j];
    }
}

// ---------------------------------------------------------------------------
// K3: stable scatter of metadata. dest = binBase[key] + blockPrefix + rank,
// where rank = #earlier tokens in this block with same key (deterministic).
__global__ __launch_bounds__(TPB) void scatter_meta(const int* __restrict__ keys,
                                                    const int* __restrict__ hist,
                                                    const int* __restrict__ binBase,
                                                    const float* __restrict__ wts,
                                                    float* __restrict__ out,
                                                    int* __restrict__ srcOfDest) {
    const int b = blockIdx.x, t = threadIdx.x;
    const int i = b * TPB + t;
    __shared__ int sk[TPB];
    const int key = keys[i];
    sk[t] = key;
    __syncthreads();
    int rank = 0;
    for (int j = 0; j < t; ++j) rank += (sk[j] == key) ? 1 : 0;

    const int dest = binBase[key] + hist[key * NBLK + b] + rank;

    out[O_KEYS + dest]      = (float)key;
    out[O_ORDER + dest]     = (float)i;
    out[O_W + dest * 2 + 0] = wts[i * 2 + 0];
    out[O_W + dest * 2 + 1] = wts[i * 2 + 1];
    srcOfDest[dest] = i;
}

// ---------------------------------------------------------------------------
// K4: bulk row gather, one workgroup per output row (16 KB). Stage through
// LDS with CDNA5 async global<->LDS DMA when available; each lane stores the
// exact bytes it loaded, so a per-wave s_wait_asynccnt suffices (no barrier).
#if __has_builtin(__builtin_amdgcn_global_load_async_to_lds_b128) && \
    __has_builtin(__builtin_amdgcn_global_store_async_from_lds_b128)
#define USE_ASYNC_LDS 1
typedef int v4i __attribute__((ext_vector_type(4)));
typedef __attribute__((address_space(1))) v4i as1_v4i;
typedef __attribute__((address_space(3))) v4i as3_v4i;
#else
#define USE_ASYNC_LDS 0
#endif

__global__ __launch_bounds__(TPB) void gather_rows(const float* __restrict__ tokens,
                                                   const int* __restrict__ srcOfDest,
                                                   float* __restrict__ outTok) {
    const int p = blockIdx.x;      // output row
    const int t = threadIdx.x;
    const int src = srcOfDest[p];
    const float* srow = tokens + (size_t)src * DIM;
    float* drow = outTok + (size_t)p * DIM;

#if USE_ASYNC_LDS
    __shared__ __align__(16) float buf[DIM];
#pragma unroll
    for (int c = 0; c < DIM / (TPB * 4); ++c) {
        const int off = c * (TPB * 4) + t * 4;       // 16B-aligned chunks
        __builtin_amdgcn_global_load_async_to_lds_b128(
            (as1_v4i*)(srow + off), (as3_v4i*)&buf[off], 0, 0);
    }
#if __has_builtin(__builtin_amdgcn_s_wait_asynccnt)
    __builtin_amdgcn_s_wait_asynccnt(0);
#else
    asm volatile("s_wait_asynccnt 0" ::: "memory");
#endif
#pragma unroll
    for (int c = 0; c < DIM / (TPB * 4); ++c) {
        const int off = c * (TPB * 4) + t * 4;
        __builtin_amdgcn_global_store_async_from_lds_b128(
            (as1_v4i*)(drow + off), (as3_v4i*)&buf[off], 0, 0);
    }
#if __has_builtin(__builtin_amdgcn_s_wait_asynccnt)
    __builtin_amdgcn_s_wait_asynccnt(0);
#else
    asm volatile("s_wait_asynccnt 0" ::: "memory");
#endif
#else
    // Fallback: plain 128-bit register copy
#pragma unroll
    for (int c = 0; c < DIM / (TPB * 4); ++c) {
        const int off = c * (TPB * 4) + t * 4;
        float4 v = *(const float4*)(srow + off);
        *(float4*)(drow + off) = v;
    }
#endif
}

// ---------------------------------------------------------------------------
extern "C" void kernel_launch(void* const* d_in, const int* in_sizes, int n_in,
                              void* d_out, int out_size, void* d_ws, size_t ws_size,
                              hipStream_t stream) {
    const float* tokens = (const float*)d_in[0];   // [NTOK, DIM] f32
    const int*   eidx   = (const int*)d_in[1];     // [NTOK, 2]  i32
    const float* ewts   = (const float*)d_in[2];   // [NTOK, 2]  f32
    float* out = (float*)d_out;

    // workspace layout (ints): keys | srcOfDest | hist[NPAT*NBLK] | binBase
    int* ws      = (int*)d_ws;
    int* wsKeys  = ws;                              // NTOK
    int* wsSrc   = ws + NTOK;                       // NTOK
    int* wsHist  = ws + 2 * NTOK;                   // NPAT * NBLK
    int* wsBase  = ws + 2 * NTOK + NPAT * NBLK;     // NPAT

    zero_i32<<<(NPAT * NBLK + TPB - 1) / TPB, TPB, 0, stream>>>(wsHist, NPAT * NBLK);
    keys_hist<<<NBLK, TPB, 0, stream>>>(eidx, wsKeys, wsHist);
    scan_bins<<<1, TPB, 0, stream>>>(wsHist, wsBase, out + O_CNT);
    scatter_meta<<<NBLK, TPB, 0, stream>>>(wsKeys, wsHist, wsBase, ewts, out, wsSrc);
    gather_rows<<<NTOK, TPB, 0, stream>>>(tokens, wsSrc, out + O_TOK);
}